// NGCF_75127567941781
// MI455X (gfx1250) — compile-verified
//
#include <hip/hip_runtime.h>
#include <hip/hip_bf16.h>

#define N_USERS 100000
#define N_ITEMS 50000
#define N_NODES (N_USERS + N_ITEMS)
#define LATENT  64

typedef float v2f __attribute__((ext_vector_type(2)));
typedef float v8f __attribute__((ext_vector_type(8)));

// ---------------------------------------------------------------------------
// SpMM scatter: side[rows[e]] += vals[e] * ego[cols[e]]   (one wave per edge,
// each lane owns a float2 of the 64-wide embedding; L2-resident atomics)
// ---------------------------------------------------------------------------
__global__ void ngcf_spmm_kernel(const int* __restrict__ rows,
                                 const int* __restrict__ cols,
                                 const float* __restrict__ vals,
                                 const float* __restrict__ ego,
                                 float* __restrict__ side,
                                 int nnz) {
    int wid  = (blockIdx.x * blockDim.x + threadIdx.x) >> 5;
    int lane = threadIdx.x & 31;
    if (wid >= nnz) return;
    int r = rows[wid];
    int c = cols[wid];
    float v = vals[wid];
    const float2* ego2 = (const float2*)ego;
    float2 m = ego2[(size_t)c * 32 + lane];
    float* dst = side + (size_t)r * LATENT + lane * 2;
    atomicAdd(dst + 0, v * m.x);
    atomicAdd(dst + 1, v * m.y);
}

// ---------------------------------------------------------------------------
// Fused NGCF layer: out = l2norm( leakyrelu( side@Wg + (ego*side)@Wb + bg+bb ) )
// Block = 128 threads (4 waves). Wave w computes the 16x16 output tile at
// column tile w; block handles a 16-row strip. fp32 WMMA 16x16x4, K-loop of 16,
// both GEMMs accumulate into one v8f accumulator (32 v_wmma per wave).
// ---------------------------------------------------------------------------
__global__ void ngcf_layer_kernel(const float* __restrict__ A,   // side  [N,64]
                                  const float* __restrict__ E,   // ego   [N,64]
                                  const float* __restrict__ Wg,  // [64,64]
                                  const float* __restrict__ bg,  // [64]
                                  const float* __restrict__ Wb,  // [64,64]
                                  const float* __restrict__ bb,  // [64]
                                  float* __restrict__ out) {     // [N,64]
    __shared__ float sh[16][65];
    __shared__ float ninv[16];

    const int lane  = threadIdx.x & 31;
    const int ntile = threadIdx.x >> 5;      // 0..3 column tile
    const int half  = lane >> 4;             // 0 or 1 (K-pair select)
    const int l15   = lane & 15;
    const int arow  = blockIdx.x * 16 + l15; // A-frag row (M = lane&15)
    const int ncol  = ntile * 16 + l15;      // B/C/D column (N = lane&15)

    const float* Arow = A + (size_t)arow * LATENT;
    const float* Erow = E + (size_t)arow * LATENT;

    v8f acc = {};
#pragma unroll
    for (int k = 0; k < 16; ++k) {
        const int kb = k * 4 + half * 2;     // this half-wave's K pair
        v2f a, a2, b, b2;
        float e0 = Erow[kb], e1v = Erow[kb + 1];
        a.x  = Arow[kb];
        a.y  = Arow[kb + 1];
        a2.x = a.x * e0;                     // (ego ⊙ side) fragment
        a2.y = a.y * e1v;
        b.x  = Wg[(size_t)kb * LATENT + ncol];
        b.y  = Wg[(size_t)(kb + 1) * LATENT + ncol];
        b2.x = Wb[(size_t)kb * LATENT + ncol];
        b2.y = Wb[(size_t)(kb + 1) * LATENT + ncol];
        acc = __builtin_amdgcn_wmma_f32_16x16x4_f32(false, a,  false, b,
                                                    (short)0, acc, false, false);
        acc = __builtin_amdgcn_wmma_f32_16x16x4_f32(false, a2, false, b2,
                                                    (short)0, acc, false, false);
    }

    // bias + leaky relu(0.2) -> LDS (C/D layout: VGPR v holds M = v + 8*half)
    const float bsum = bg[ncol] + bb[ncol];
#pragma unroll
    for (int v = 0; v < 8; ++v) {
        int m = v + half * 8;
        float z = acc[v] + bsum;
        z = z > 0.0f ? z : 0.2f * z;
        sh[m][ncol] = z;
    }
    __syncthreads();

    // row L2 norms (16 rows of 64)
    if (threadIdx.x < 16) {
        float s = 0.0f;
        for (int c = 0; c < 64; ++c) { float t = sh[threadIdx.x][c]; s += t * t; }
        ninv[threadIdx.x] = 1.0f / fmaxf(sqrtf(s), 1e-12f);
    }
    __syncthreads();

    // normalized writeback: 128 threads x 8 contiguous elements
    const int base = threadIdx.x * 8;
    const int m = base >> 6;
    const int c0 = base & 63;
    const float inv = ninv[m];
    float* orow = out + ((size_t)blockIdx.x * 16 + m) * LATENT + c0;
#pragma unroll
    for (int j = 0; j < 8; ++j) orow[j] = sh[m][c0 + j] * inv;
}

// ---------------------------------------------------------------------------
// gamma[b] = sum_k dot(emb_k[user], emb_k[N_USERS+item]) for k in {x, e1, e2}
// One wave per batch element, shuffle reduction (wave32).
// ---------------------------------------------------------------------------
__global__ void ngcf_gamma_kernel(const float* __restrict__ x,
                                  const float* __restrict__ e1,
                                  const float* __restrict__ e2,
                                  const int* __restrict__ users,
                                  const int* __restrict__ items,
                                  float* __restrict__ out,
                                  int batch) {
    int wid  = (blockIdx.x * blockDim.x + threadIdx.x) >> 5;
    int lane = threadIdx.x & 31;
    if (wid >= batch) return;
    size_t uo = (size_t)users[wid] * LATENT;
    size_t io = (size_t)(N_USERS + items[wid]) * LATENT;
    float s = 0.0f;
    s += x [uo + lane] * x [io + lane] + x [uo + lane + 32] * x [io + lane + 32];
    s += e1[uo + lane] * e1[io + lane] + e1[uo + lane + 32] * e1[io + lane + 32];
    s += e2[uo + lane] * e2[io + lane] + e2[uo + lane + 32] * e2[io + lane + 32];
#pragma unroll
    for (int off = 16; off > 0; off >>= 1) s += __shfl_xor(s, off, 32);
    if (lane == 0) out[wid] = s;
}

extern "C" void kernel_launch(void* const* d_in, const int* in_sizes, int n_in,
                              void* d_out, int out_size, void* d_ws, size_t ws_size,
                              hipStream_t stream) {
    // setup_inputs() dict order:
    const int*   rows  = (const int*)  d_in[0];
    const int*   cols  = (const int*)  d_in[1];
    const float* vals  = (const float*)d_in[2];
    const float* x     = (const float*)d_in[3];
    const int*   users = (const int*)  d_in[4];
    const int*   items = (const int*)  d_in[5];
    const float* Wg0   = (const float*)d_in[6];
    const float* bg0   = (const float*)d_in[7];
    const float* Wb0   = (const float*)d_in[8];
    const float* bb0   = (const float*)d_in[9];
    const float* Wg1   = (const float*)d_in[10];
    const float* bg1   = (const float*)d_in[11];
    const float* Wb1   = (const float*)d_in[12];
    const float* bb1   = (const float*)d_in[13];
    const int nnz   = in_sizes[0];
    const int batch = in_sizes[4];

    const size_t embN = (size_t)N_NODES * LATENT;      // 9.6M floats
    float* side = (float*)d_ws;
    float* e1   = side + embN;
    float* e2   = e1 + embN;

    const int spmm_blocks  = (nnz + 7) / 8;            // 8 waves (edges) / block
    const int layer_blocks = N_NODES / 16;             // 9375, exact
    const int gamma_blocks = (batch * 32 + 255) / 256;

    // ---- layer 1 ----
    hipMemsetAsync(side, 0, embN * sizeof(float), stream);
    ngcf_spmm_kernel<<<spmm_blocks, 256, 0, stream>>>(rows, cols, vals, x, side, nnz);
    ngcf_layer_kernel<<<layer_blocks, 128, 0, stream>>>(side, x, Wg0, bg0, Wb0, bb0, e1);

    // ---- layer 2 ----
    hipMemsetAsync(side, 0, embN * sizeof(float), stream);
    ngcf_spmm_kernel<<<spmm_blocks, 256, 0, stream>>>(rows, cols, vals, e1, side, nnz);
    ngcf_layer_kernel<<<layer_blocks, 128, 0, stream>>>(side, e1, Wg1, bg1, Wb1, bb1, e2);

    // ---- batched dot products ----
    ngcf_gamma_kernel<<<gamma_blocks, 256, 0, stream>>>(x, e1, e2, users, items,
                                                        (float*)d_out, batch);
}